// FNO4d_45964740002369
// MI455X (gfx1250) — compile-verified
//
#include <hip/hip_runtime.h>
#include <math.h>

// FNO-4D forward for MI455X (gfx1250), fp32 end-to-end.
// All DFT stages + pointwise/head GEMMs run on V_WMMA_F32_16X16X4_F32; spectral mix on VALU.
// Weight tables staged into LDS with TENSOR_LOAD_TO_LDS (TDM) + s_wait_tensorcnt.
// Workspace layout (floats): h[20971520] h1[20971520] bufA[15728640] bufB[15728640]
//   -> 293,601,280 bytes of d_ws used.

typedef float v2f __attribute__((ext_vector_type(2)));
typedef float v8f __attribute__((ext_vector_type(8)));
typedef unsigned int u32x4 __attribute__((ext_vector_type(4)));
typedef int i32x4 __attribute__((ext_vector_type(4)));
typedef int i32x8 __attribute__((ext_vector_type(8)));

#define WMMA_ACC(a, b, c) \
  __builtin_amdgcn_wmma_f32_16x16x4_f32(false, (a), false, (b), (short)0, (c), false, false)

#if __has_builtin(__builtin_amdgcn_tensor_load_to_lds)
#define HAVE_TDM 1
#else
#define HAVE_TDM 0
#endif

__device__ __forceinline__ float gelu_f(float x) {
  return 0.5f * x * (1.0f + erff(x * 0.70710678118654752440f));
}

#if HAVE_TDM
// 1-D TDM copy: nfloats contiguous floats from gsrc into LDS at byte offset lds_off.
// D# layout per CDNA5 ISA 8.3/8.4: group0 = {count|flags, lds_addr, gaddr_lo, gaddr_hi|type=2},
// group1 = {mask|data_size, tensor_dim0, tensor_dim1, tile_dim0, tile_dim1, strides...}.
__device__ __forceinline__ void tdm_load_1d(const float* gsrc, unsigned lds_off, int nfloats) {
  unsigned long long ga = (unsigned long long)(const void*)gsrc;
  u32x4 g0;
  g0[0] = 1u;                                            // count=1 (valid user D#)
  g0[1] = lds_off;                                       // lds_addr (bytes)
  g0[2] = (unsigned)(ga & 0xFFFFFFFFu);                  // global_addr[31:0]
  g0[3] = (unsigned)((ga >> 32) & 0x01FFFFFFu) | (2u << 30);  // addr[56:32] | type=2
  i32x8 g1;
  g1[0] = (2 << 16);                                     // workgroup_mask=0, data_size=4B
  g1[1] = (int)(((unsigned)nfloats & 0xFFFFu) << 16);    // tensor_dim0[15:0] @bits63:48
  g1[2] = (int)((((unsigned)nfloats >> 16) & 0xFFFFu) | (1u << 16)); // dim0 hi | tensor_dim1=1
  g1[3] = (int)(((unsigned)nfloats & 0xFFFFu) << 16);    // tile_dim0 @bits127:112
  g1[4] = 1;                                             // tile_dim1=1
  g1[5] = nfloats;                                       // tensor_dim0_stride lo
  g1[6] = 0;
  g1[7] = 0;
  i32x4 z4 = {0, 0, 0, 0};
#if defined(__clang_major__) && (__clang_major__ >= 23)
  i32x8 z8 = {0, 0, 0, 0, 0, 0, 0, 0};
  __builtin_amdgcn_tensor_load_to_lds(g0, g1, z4, z4, z8, 0);
#else
  __builtin_amdgcn_tensor_load_to_lds(g0, g1, z4, z4, 0);
#endif
}
#endif

// ---------------------------------------------------------------- lift 5->20
__global__ void k_lift(const float* __restrict__ x, const float* __restrict__ w,
                       const float* __restrict__ bia, float* __restrict__ h) {
  int p = blockIdx.x * blockDim.x + threadIdx.x;
  if (p >= 1048576) return;
  int b = p >> 19;               // / 524288
  int pos = p & 524287;
  float xi[5];
#pragma unroll
  for (int i = 0; i < 5; ++i) xi[i] = x[(size_t)p * 5 + i];
#pragma unroll
  for (int c = 0; c < 20; ++c) {
    float acc = bia[c];
#pragma unroll
    for (int i = 0; i < 5; ++i) acc += xi[i] * w[i * 20 + c];
    h[(size_t)(b * 20 + c) * 524288 + pos] = acc;
  }
}

// ------------------------------------------- forward T-stage: 16 real -> 6 cplx
// rows = bc*32768 + xyz (1,310,720). in: h[row*16 + t]. out: float[row*12 + (2q|re,im)]
__global__ void k_ft(const float* __restrict__ h, float* __restrict__ out) {
  __shared__ float Tb[256];
  int tid = threadIdx.x;
  for (int idx = tid; idx < 256; idx += blockDim.x) {
    int t = idx >> 4, n = idx & 15;
    float v = 0.f;
    if (n < 12) {
      int q = n >> 1;
      float th = 6.28318530717958647692f * (float)(q * t) / 16.f;
      v = (n & 1) ? -__sinf(th) : __cosf(th);   // e^{-i th}
    }
    Tb[idx] = v;
  }
  __syncthreads();
  int lane = tid & 31, wave = tid >> 5;
  long tile = (long)blockIdx.x * (blockDim.x >> 5) + wave;
  if (tile >= 81920) return;
  int r = lane & 15, hi = lane >> 4;
  long row0 = tile * 16;
  v8f c = {};
#pragma unroll
  for (int k0 = 0; k0 < 16; k0 += 4) {
    int kk = k0 + 2 * hi;
    float2 av = *(const float2*)(h + (row0 + r) * 16 + kk);
    v2f a = {av.x, av.y};
    v2f bb = {Tb[kk * 16 + r], Tb[(kk + 1) * 16 + r]};
    c = WMMA_ACC(a, bb, c);
  }
  int n = lane & 15;
  if (n < 12) {
#pragma unroll
    for (int v = 0; v < 8; ++v) {
      long row = row0 + v + 8 * hi;
      out[row * 12 + n] = c[v];
    }
  }
}

// ----------------------- generic complex DFT stage along one axis (WMMA f32)
// layout: in[(o*Lk + k)*I + i], out[(o*Ln + n)*I + i]
// forward (inverse==0): Lk=32 spatial -> Ln=16 freqs {0..7,24..31}, e^{-i}
// inverse (inverse==1): Lk=16 freqs -> Ln=32 spatial, e^{+i}/32
__global__ void k_cdft(const float2* __restrict__ in, float2* __restrict__ out,
                       int O, int I, int Lk, int Ln, int inverse) {
  __shared__ float Br[1024], Bi[1024];
  int tid = threadIdx.x;
  int tot = Lk * Ln;
  for (int idx = tid; idx < tot; idx += blockDim.x) {
    int k = idx / Ln, n = idx - k * Ln;
    float re, im;
    if (!inverse) {
      int f = (n < 8) ? n : n + 16;
      float th = 6.28318530717958647692f * (float)(f * k) / (float)Lk;
      re = __cosf(th); im = -__sinf(th);
    } else {
      int f = (k < 8) ? k : k + 16;
      float th = 6.28318530717958647692f * (float)(f * n) / (float)Ln;
      re = __cosf(th) * (1.f / 32.f); im = __sinf(th) * (1.f / 32.f);
    }
    Br[idx] = re; Bi[idx] = im;
  }
  __syncthreads();
  int lane = tid & 31, wave = tid >> 5;
  int ntn = Ln >> 4;
  long R = (long)O * I;
  long units = (R >> 4) * ntn;
  long unit = (long)blockIdx.x * (blockDim.x >> 5) + wave;
  if (unit >= units) return;
  long mt = unit / ntn;
  int nt = (int)(unit - mt * ntn);
  long row0 = mt << 4;
  int l15 = lane & 15, hi = lane >> 4;
  int n = nt * 16 + l15;
  long rA = row0 + l15;
  int oA = (int)(rA / I), iA = (int)(rA - (long)oA * I);
  const float2* base = in + (size_t)oA * Lk * I + iA;
  v8f cr = {}, ci = {};
  for (int k0 = 0; k0 < Lk; k0 += 4) {
    int kk = k0 + 2 * hi;
    float2 e0 = base[(size_t)kk * I];
    float2 e1 = base[(size_t)(kk + 1) * I];
    v2f ar = {e0.x, e1.x};
    v2f ai = {e0.y, e1.y};
    v2f ain = {-e0.y, -e1.y};            // f32 WMMA has no A-neg modifier
    v2f br = {Br[kk * Ln + n], Br[(kk + 1) * Ln + n]};
    v2f bi = {Bi[kk * Ln + n], Bi[(kk + 1) * Ln + n]};
    cr = WMMA_ACC(ar, br, cr);
    cr = WMMA_ACC(ain, bi, cr);          // Cr -= Ai*Bi
    ci = WMMA_ACC(ar, bi, ci);
    ci = WMMA_ACC(ai, br, ci);
  }
#pragma unroll
  for (int v = 0; v < 8; ++v) {
    long row = row0 + v + 8 * hi;
    int o = (int)(row / I), ii = (int)(row - (long)o * I);
    out[((size_t)o * Ln + n) * I + ii] = make_float2(cr[v], ci[v]);
  }
}

// ------------------------- per-mode complex 20x20 channel mix (VALU; M=2 only)
__global__ void k_specmul(const float2* __restrict__ in, float2* __restrict__ out,
                          const float* __restrict__ sw) {
  int t = blockIdx.x * blockDim.x + threadIdx.x;
  if (t >= 983040) return;
  int o = t % 20;
  int rem = t / 20;
  int m = rem % 24576;
  int b = rem / 24576;
  int kt = m % 6;
  int kz = (m / 6) & 15;
  int ky = (m / 96) & 15;
  int kx = m / 1536;
  int oct = ((kx >> 3) << 2) | ((ky >> 3) << 1) | (kz >> 3);
  int mm = (((kx & 7) * 8 + (ky & 7)) * 8 + (kz & 7)) * 6 + kt;
  // w index: ((((oct*20 + i)*20 + o))*3072 + mm)*2 ; per-i stride 122880 floats
  const float* wp = sw + ((size_t)(oct * 20) * 20 + o) * 3072u * 2u + (size_t)mm * 2u;
  const float2* ip = in + (size_t)b * 20 * 24576 + m;
  __builtin_prefetch(wp, 0, 1);
  float accr = 0.f, acci = 0.f;
#pragma unroll 4
  for (int i = 0; i < 20; ++i) {
    float2 xv = ip[(size_t)i * 24576];
    float wr = wp[(size_t)i * 122880];
    float wi = wp[(size_t)i * 122880 + 1];
    accr += xv.x * wr - xv.y * wi;
    acci += xv.x * wi + xv.y * wr;
  }
  out[((size_t)b * 20 + o) * 24576 + m] = make_float2(accr, acci);
}

// ------------------------------ inverse T-stage: 6 cplx bins -> 16 real (irfft)
// in: float[row*12 + (2q|re,im)], out: h1[row*16 + t]; scale 1/16, bins1..5 x2
__global__ void k_it(const float* __restrict__ in, float* __restrict__ h1) {
  __shared__ float Tb[192];
  int tid = threadIdx.x;
  for (int idx = tid; idx < 192; idx += blockDim.x) {
    int k = idx >> 4, n = idx & 15;
    int q = k >> 1;
    float cq = (q == 0) ? 1.f : 2.f;
    float th = 6.28318530717958647692f * (float)(q * n) / 16.f;
    float v = (k & 1) ? (-cq * __sinf(th)) : (cq * __cosf(th));
    Tb[idx] = v * (1.f / 16.f);
  }
  __syncthreads();
  int lane = tid & 31, wave = tid >> 5;
  long tile = (long)blockIdx.x * (blockDim.x >> 5) + wave;
  if (tile >= 81920) return;
  int r = lane & 15, hi = lane >> 4;
  long row0 = tile * 16;
  v8f c = {};
#pragma unroll
  for (int k0 = 0; k0 < 12; k0 += 4) {
    int kk = k0 + 2 * hi;
    float2 av = *(const float2*)(in + (row0 + r) * 12 + kk);
    v2f a = {av.x, av.y};
    v2f bb = {Tb[kk * 16 + r], Tb[(kk + 1) * 16 + r]};
    c = WMMA_ACC(a, bb, c);
  }
#pragma unroll
  for (int v = 0; v < 8; ++v) {
    long row = row0 + v + 8 * hi;
    h1[row * 16 + (lane & 15)] = c[v];
  }
}

// ---------------- h = maybe_gelu(h1 + h @ ww^T + wb), in place on h (WMMA)
// ww (400 floats) + wb (20 floats) staged into LDS by the Tensor Data Mover.
__global__ void k_pointwise(float* __restrict__ h, const float* __restrict__ h1,
                            const float* __restrict__ ww, const float* __restrict__ wb,
                            int apply_gelu) {
  __shared__ float S[420];
  const float* Wl = S;
  const float* Bv = S + 400;
  int tid = threadIdx.x;
#if HAVE_TDM
  if ((tid >> 5) == 0) {
    unsigned base = (unsigned)(size_t)(const void*)S;
    tdm_load_1d(ww, base, 400);
    tdm_load_1d(wb, base + 400u * 4u, 20);
    __builtin_amdgcn_s_wait_tensorcnt(0);
  }
#else
  for (int i = tid; i < 400; i += blockDim.x) S[i] = ww[i];
  if (tid < 20) S[400 + tid] = wb[tid];
#endif
  __syncthreads();
  int lane = tid & 31, wave = tid >> 5;
  long tile = (long)blockIdx.x * (blockDim.x >> 5) + wave;
  if (tile >= 65536) return;
  long p0 = tile * 16;
  int b = (int)(p0 >> 19);
  long pos0 = p0 & 524287;
  int r = lane & 15, hi = lane >> 4;
  int n1 = 16 + r;
  float* hb = h + (size_t)b * 20 * 524288;
  v8f c0 = {}, c1 = {};
#pragma unroll
  for (int k0 = 0; k0 < 20; k0 += 4) {
    int kk = k0 + 2 * hi;
    v2f a = {hb[(size_t)kk * 524288 + pos0 + r],
             hb[(size_t)(kk + 1) * 524288 + pos0 + r]};
    v2f b0 = {Wl[r * 20 + kk], Wl[r * 20 + kk + 1]};           // B[k][n]=ww[n*20+k]
    v2f b1 = {(n1 < 20) ? Wl[n1 * 20 + kk] : 0.f,
              (n1 < 20) ? Wl[n1 * 20 + kk + 1] : 0.f};
    c0 = WMMA_ACC(a, b0, c0);
    c1 = WMMA_ACC(a, b1, c1);
  }
#pragma unroll
  for (int v = 0; v < 8; ++v) {
    long pos = pos0 + v + 8 * hi;
    {
      int o = r;
      float val = c0[v] + h1[(size_t)(b * 20 + o) * 524288 + pos] + Bv[o];
      if (apply_gelu) val = gelu_f(val);
      hb[(size_t)o * 524288 + pos] = val;
    }
    if (n1 < 20) {
      int o = n1;
      float val = c1[v] + h1[(size_t)(b * 20 + o) * 524288 + pos] + Bv[o];
      if (apply_gelu) val = gelu_f(val);
      hb[(size_t)o * 524288 + pos] = val;
    }
  }
}

// ------------- fused head: out = gelu(h@fc1 + b1) @ fc2 + b2  (WMMA + shfl)
// fc1_w/fc1_b/fc2_w (2816 floats) staged into LDS by the Tensor Data Mover.
__global__ void k_fchead(const float* __restrict__ h, const float* __restrict__ w1,
                         const float* __restrict__ b1, const float* __restrict__ w2,
                         const float* __restrict__ b2, float* __restrict__ out) {
  __shared__ float S[2816];
  const float* W1 = S;
  const float* B1 = S + 2560;
  const float* W2 = S + 2688;
  int tid = threadIdx.x;
#if HAVE_TDM
  if ((tid >> 5) == 0) {
    unsigned base = (unsigned)(size_t)(const void*)S;
    tdm_load_1d(w1, base, 2560);
    tdm_load_1d(b1, base + 2560u * 4u, 128);
    tdm_load_1d(w2, base + 2688u * 4u, 128);
    __builtin_amdgcn_s_wait_tensorcnt(0);
  }
#else
  for (int i = tid; i < 2560; i += blockDim.x) S[i] = w1[i];
  for (int i = tid; i < 128; i += blockDim.x) { S[2560 + i] = b1[i]; S[2688 + i] = w2[i]; }
#endif
  __syncthreads();
  int lane = tid & 31, wave = tid >> 5;
  long tile = (long)blockIdx.x * (blockDim.x >> 5) + wave;
  if (tile >= 65536) return;
  long p0 = tile * 16;
  int b = (int)(p0 >> 19);
  long pos0 = p0 & 524287;
  int r = lane & 15, hi = lane >> 4;
  const float* hb = h + (size_t)b * 20 * 524288;
  v8f c[8] = {};
#pragma unroll
  for (int k0 = 0; k0 < 20; k0 += 4) {
    int kk = k0 + 2 * hi;
    v2f a = {hb[(size_t)kk * 524288 + pos0 + r],
             hb[(size_t)(kk + 1) * 524288 + pos0 + r]};
#pragma unroll
    for (int nt = 0; nt < 8; ++nt) {
      int n = nt * 16 + r;
      v2f bb = {W1[kk * 128 + n], W1[(kk + 1) * 128 + n]};
      c[nt] = WMMA_ACC(a, bb, c[nt]);
    }
  }
  float acc[8];
#pragma unroll
  for (int v = 0; v < 8; ++v) acc[v] = 0.f;
#pragma unroll
  for (int nt = 0; nt < 8; ++nt) {
    int n = nt * 16 + r;
    float bias = B1[n], w2v = W2[n];
#pragma unroll
    for (int v = 0; v < 8; ++v) acc[v] += gelu_f(c[nt][v] + bias) * w2v;
  }
#pragma unroll
  for (int v = 0; v < 8; ++v) {
#pragma unroll
    for (int off = 1; off < 16; off <<= 1)
      acc[v] += __shfl_xor(acc[v], off, 16);
  }
  if (r == 0) {
    float bo = b2[0];
#pragma unroll
    for (int v = 0; v < 8; ++v)
      out[(size_t)b * 524288 + pos0 + v + 8 * hi] = acc[v] + bo;
  }
}

extern "C" void kernel_launch(void* const* d_in, const int* in_sizes, int n_in,
                              void* d_out, int out_size, void* d_ws, size_t ws_size,
                              hipStream_t stream) {
  (void)in_sizes; (void)n_in; (void)out_size; (void)ws_size;
  const float* x    = (const float*)d_in[0];
  const float* fc0w = (const float*)d_in[1];
  const float* fc0b = (const float*)d_in[2];
  const float* sw   = (const float*)d_in[3];
  const float* ww   = (const float*)d_in[4];
  const float* wb   = (const float*)d_in[5];
  const float* fc1w = (const float*)d_in[6];
  const float* fc1b = (const float*)d_in[7];
  const float* fc2w = (const float*)d_in[8];
  const float* fc2b = (const float*)d_in[9];
  float* out = (float*)d_out;

  float* f    = (float*)d_ws;
  float* h    = f;                       // [40][524288]
  float* h1   = h + 20971520ull;         // [40][524288]
  float* bufA = h1 + 20971520ull;        // 15,728,640 floats
  float* bufB = bufA + 15728640ull;      // 15,728,640 floats

  k_lift<<<4096, 256, 0, stream>>>(x, fc0w, fc0b, h);

  for (int l = 0; l < 4; ++l) {
    // forward truncated DFT: T, Z, Y, X
    k_ft<<<10240, 256, 0, stream>>>(h, bufA);
    k_cdft<<<1920, 256, 0, stream>>>((const float2*)bufA, (float2*)bufB, 40960, 6,   32, 16, 0);
    k_cdft<<< 960, 256, 0, stream>>>((const float2*)bufB, (float2*)bufA, 1280,  96,  32, 16, 0);
    k_cdft<<< 480, 256, 0, stream>>>((const float2*)bufA, (float2*)bufB, 40,    1536,32, 16, 0);
    // per-mode complex channel mix
    k_specmul<<<3840, 256, 0, stream>>>((const float2*)bufB, (float2*)bufA,
                                        sw + (size_t)l * 19660800ull);
    // inverse DFT: X, Y, Z, T (1/32 each, 1/16 in T)
    k_cdft<<< 960, 256, 0, stream>>>((const float2*)bufA, (float2*)bufB, 40,    1536,16, 32, 1);
    k_cdft<<<1920, 256, 0, stream>>>((const float2*)bufB, (float2*)bufA, 1280,  96,  16, 32, 1);
    k_cdft<<<3840, 256, 0, stream>>>((const float2*)bufA, (float2*)bufB, 40960, 6,   16, 32, 1);
    k_it<<<10240, 256, 0, stream>>>(bufB, h1);
    // h = maybe_gelu(h1 + h @ ww^T + wb), in place
    k_pointwise<<<8192, 256, 0, stream>>>(h, h1, ww + l * 400, wb + l * 20, (l < 3) ? 1 : 0);
  }

  k_fchead<<<8192, 256, 0, stream>>>(h, fc1w, fc1b, fc2w, fc2b, out);
}